// RotateMyLayer_5136780886688
// MI455X (gfx1250) — compile-verified
//
#include <hip/hip_runtime.h>

// Problem constants (match reference)
#define OUT_H     8
#define MAX_W     384
#define NUM_BOXES 256
#define FH        160
#define FW        160
#define FC        32

typedef float f4 __attribute__((ext_vector_type(4)));

// 32B per sample point: 4 bilinear weights (validity mask folded in) and
// 4 corner byte-offsets into the feature map. 16B-aligned so phase-2 reads
// lower to two ds_load_b128 (8 threads read the same address -> LDS broadcast).
struct __align__(16) PointData {
    float wa, wb, wc, wd;
    int   offA, offB, offC, offD;
};

// Grid: (MAX_W/32, OUT_H, NUM_BOXES), block = 256 threads (8 wave32).
// Phase 1: wave 0 computes per-point weights/offsets for the 32-column tile
//          (division/floor/clamp chain executed once per point, not 8x).
// Phase 2: threadIdx.x = { j_in_tile[4:0], chan_group[2:0] } -> 4 coalesced
//          128b corner gathers + one contiguous 4KB NT b128 store per block.
__global__ __launch_bounds__(256)
void stn_bilinear_kernel(const float* __restrict__ feat,   // (8,160,160,32) f32
                         const float* __restrict__ theta,  // (256,6) f32
                         const int*   __restrict__ box,    // (256,2) i32 [img,width]
                         float* __restrict__ out)          // (256,8,384,32) f32
{
    __shared__ PointData pd[32];

    const int i     = blockIdx.y;        // output row (uniform)
    const int n     = blockIdx.z;        // box index  (uniform)
    const int jbase = blockIdx.x * 32;   // first column of this tile

    if (threadIdx.x < 32) {              // exactly wave 0
        const int p = threadIdx.x;
        const int j = jbase + p;

        // Uniform per-box data -> scalar (SMEM) loads
        const int img = box[2 * n + 0];
        const int bw  = box[2 * n + 1];
        const float t0 = theta[6 * n + 0];
        const float t1 = theta[6 * n + 1];
        const float t2 = theta[6 * n + 2];
        const float t3 = theta[6 * n + 3];
        const float t4 = theta[6 * n + 4];
        const float t5 = theta[6 * n + 5];

        // Normalized grid (same op order as reference: exact IEEE division)
        const float x_t = -1.0f + 2.0f * (float)j / ((float)bw - 1.0f);
        const float y_t = -1.0f + 2.0f * (float)i / (float)(OUT_H - 1);

        // Affine transform -> pixel coords
        const float xs = t0 * x_t + t1 * y_t + t2;
        const float ys = t3 * x_t + t4 * y_t + t5;
        const float x  = (xs + 1.0f) * (0.5f * (float)FW);
        const float y  = (ys + 1.0f) * (0.5f * (float)FH);

        // Corner indices, clamped
        const int x0 = (int)floorf(x);
        const int y0 = (int)floorf(y);
        const int x0c = min(max(x0, 0),     FW - 1);
        const int x1c = min(max(x0 + 1, 0), FW - 1);
        const int y0c = min(max(y0, 0),     FH - 1);
        const int y1c = min(max(y0 + 1, 0), FH - 1);

        // Bilinear weights from clamped coords, validity mask folded in
        const float x0f = (float)x0c, x1f = (float)x1c;
        const float y0f = (float)y0c, y1f = (float)y1c;
        const float m   = (j < bw) ? 1.0f : 0.0f;

        PointData d;
        d.wa = m * ((x1f - x) * (y1f - y));
        d.wb = m * ((x1f - x) * (y  - y0f));
        d.wc = m * ((x  - x0f) * (y1f - y));
        d.wd = m * ((x  - x0f) * (y  - y0f));

        const int rowA = (img * FH + y0c) * FW;   // y0 row
        const int rowB = (img * FH + y1c) * FW;   // y1 row
        d.offA = (rowA + x0c) * (FC * 4);         // byte offsets
        d.offB = (rowB + x0c) * (FC * 4);
        d.offC = (rowA + x1c) * (FC * 4);
        d.offD = (rowB + x1c) * (FC * 4);
        pd[p] = d;
    }
    __syncthreads();

    const int g  = threadIdx.x & 7;      // channel group (4 floats)
    const int jl = threadIdx.x >> 3;     // column within tile

    // 8 threads per point read the same 32B record -> LDS broadcast
    const PointData d = pd[jl];

    const char* fb = (const char*)feat;
    const int   cg = g * 16;             // channel-group byte offset

    // Gather 4 corners, 4 channels each (L2-resident feature map, RT hint)
    const f4 Ia = *(const f4*)(fb + d.offA + cg);
    const f4 Ib = *(const f4*)(fb + d.offB + cg);
    const f4 Ic = *(const f4*)(fb + d.offC + cg);
    const f4 Id = *(const f4*)(fb + d.offD + cg);

    f4 r;
    r.x = d.wa * Ia.x + d.wb * Ib.x + d.wc * Ic.x + d.wd * Id.x;
    r.y = d.wa * Ia.y + d.wb * Ib.y + d.wc * Ic.y + d.wd * Id.y;
    r.z = d.wa * Ia.z + d.wb * Ib.z + d.wc * Ic.z + d.wd * Id.z;
    r.w = d.wa * Ia.w + d.wb * Ib.w + d.wc * Ic.w + d.wd * Id.w;

    // Streamed-once 100MB output: non-temporal 128-bit store (gfx1250 TH=NT)
    const int oidx = ((n * OUT_H + i) * MAX_W + (jbase + jl)) * FC + g * 4;
    __builtin_nontemporal_store(r, (f4*)(out + oidx));
}

extern "C" void kernel_launch(void* const* d_in, const int* in_sizes, int n_in,
                              void* d_out, int out_size, void* d_ws, size_t ws_size,
                              hipStream_t stream) {
    const float* feat  = (const float*)d_in[0];   // shared_features
    const float* theta = (const float*)d_in[1];   // transform_matrix
    const int*   box   = (const int*)d_in[2];     // box_info
    float*       out   = (float*)d_out;

    dim3 grid(MAX_W / 32, OUT_H, NUM_BOXES);      // (12, 8, 256)
    dim3 block(256, 1, 1);
    stn_bilinear_kernel<<<grid, block, 0, stream>>>(feat, theta, box, out);
}